// IslandEncoder_21543555957431
// MI455X (gfx1250) — compile-verified
//
#include <hip/hip_runtime.h>
#include <hip/hip_bf16.h>
#include <math.h>

// ---------------------------------------------------------------------------
// GraphSAGE-style GNN for MI455X (gfx1250, wave32).
//  - GEMMs via V_WMMA_F32_16X16X4_F32 with W staged in LDS as packed K-pairs
//    (one ds_load_b64 per B fragment; 8 loads then 8 WMMAs per k-step)
//  - Edge mean-aggregation via wave-per-edge float4 gather + f32 L2 atomics
//  - Graph pooling via atomic sum + monotone-bits float atomic max
// ---------------------------------------------------------------------------

typedef float v2f __attribute__((ext_vector_type(2)));
typedef float v8f __attribute__((ext_vector_type(8)));

#define D_HID 128
#define D_GFT 32
#define NLAYER 3
#define Z_DIM (2 * D_HID + D_GFT)   // 288

// ---------------- LDS weight staging ----------------------------------------
// lw[pair * 128 + col] = { W[2*pair][col], W[2*pair+1][col] }
// -> B fragment for WMMA k-step k0, lane group g is a single 8-byte LDS load.
__device__ __forceinline__ void stage_W_lds(float2* __restrict__ lw,
                                            const float* __restrict__ W, int K)
{
  const int total = (K / 2) * D_HID;
  for (int i = threadIdx.x; i < total; i += blockDim.x) {
    const int pair = i >> 7;       // / 128
    const int col  = i & 127;
    float2 v;
    v.x = W[(size_t)(2 * pair)     * D_HID + col];
    v.y = W[(size_t)(2 * pair + 1) * D_HID + col];
    lw[i] = v;
  }
}

// ---------------- WMMA strip GEMM helpers ----------------------------------
// One wave computes a 16-row x 128-col strip of O = A(16xK) * W(Kx128).
// A layout (16x4 f32): lane L holds row m=L%16, K = 2*(L/16)+v  (v = vgpr 0/1)
// B layout (4x16 f32): lane L holds col n=L%16, K = 2*(L/16)+v
// C/D layout (16x16):  vgpr r, lane L -> row r+8*(L/16), col L%16

__device__ __forceinline__ void gemm_strip_accum(
    const float* __restrict__ A, int lda, int row0, int nrows, int K,
    const float2* __restrict__ lw,   // packed weights in LDS
    float ascale, v8f acc[8])
{
  const int lane = threadIdx.x & 31;
  const int m    = lane & 15;
  const int g    = lane >> 4;
  int row = row0 + m;
  if (row >= nrows) row = nrows - 1;   // clamp (tail-safe; N is a multiple of 16)
  const float* arow = A + (size_t)row * lda;

  for (int k0 = 0; k0 < K; k0 += 4) {
    v2f a = *(const v2f*)(arow + k0 + 2 * g);   // one global b64
    a.x *= ascale;
    a.y *= ascale;
    const float2* wb = lw + (size_t)((k0 >> 1) + g) * D_HID + m;
    v2f b[8];
#pragma unroll
    for (int t = 0; t < 8; ++t) {               // 8 x ds_load_b64, then...
      const float2 w = wb[t * 16];
      b[t].x = w.x;
      b[t].y = w.y;
    }
#pragma unroll
    for (int t = 0; t < 8; ++t) {               // ...8 back-to-back WMMAs
      acc[t] = __builtin_amdgcn_wmma_f32_16x16x4_f32(
          false, a, false, b[t], (short)0, acc[t], false, false);
    }
  }
}

__device__ __forceinline__ void store_strip_relu(
    float* __restrict__ O, int ldo, int row0, int nrows,
    const float* __restrict__ bias, v8f acc[8])
{
  const int lane = threadIdx.x & 31;
  const int n    = lane & 15;
  const int g    = lane >> 4;
#pragma unroll
  for (int t = 0; t < 8; ++t) {
#pragma unroll
    for (int r = 0; r < 8; ++r) {
      const int row = row0 + r + 8 * g;
      if (row < nrows) {
        const int col = t * 16 + n;
        float v = acc[t][r] + bias[col];
        O[(size_t)row * ldo + col] = v > 0.0f ? v : 0.0f;
      }
    }
  }
}

// ---------------- kernels ---------------------------------------------------

__global__ void fill_kernel(float* __restrict__ p, float v, size_t n) {
  size_t i = (size_t)blockIdx.x * blockDim.x + threadIdx.x;
  if (i < n) p[i] = v;
}

__global__ void deg_kernel(const int* __restrict__ dst, float* __restrict__ deg, int E) {
  int e = blockIdx.x * blockDim.x + threadIdx.x;
  if (e < E) atomicAdd(&deg[dst[e]], 1.0f);
}

// O = relu(A @ W + bias);  A: nrows x K (lda), W: K x 128, O: nrows x 128 (ldo)
template <int K>
__global__ void gemm_relu_kernel(const float* __restrict__ A, int lda,
                                 const float* __restrict__ W,
                                 const float* __restrict__ bias,
                                 float* __restrict__ O, int ldo, int nrows)
{
  __shared__ float2 lw[(K / 2) * D_HID];       // K=128 -> 64KB, K=288 -> 144KB
  stage_W_lds(lw, W, K);
  __syncthreads();

  const int wave = (blockIdx.x * blockDim.x + threadIdx.x) >> 5;
  const int row0 = wave * 16;
  if (row0 >= nrows) return;                   // wave-uniform: EXEC all-ones
  v8f acc[8];
#pragma unroll
  for (int i = 0; i < 8; ++i)
#pragma unroll
    for (int j = 0; j < 8; ++j) acc[i][j] = 0.0f;
  gemm_strip_accum(A, lda, row0, nrows, K, lw, 1.0f, acc);
  store_strip_relu(O, ldo, row0, nrows, bias, acc);
}

// hout = relu((agg/max(deg,1)) @ Wl + h @ Wr + bl)   (both GEMMs fused)
__global__ void sage_combine_kernel(const float* __restrict__ h,
                                    const float* __restrict__ agg,
                                    const float* __restrict__ deg,
                                    const float* __restrict__ Wl,
                                    const float* __restrict__ bl,
                                    const float* __restrict__ Wr,
                                    float* __restrict__ hout, int nrows)
{
  __shared__ float2 lw[D_HID * D_HID];         // two 64KB halves = 128KB
  stage_W_lds(lw, Wl, D_HID);
  stage_W_lds(lw + (D_HID / 2) * D_HID, Wr, D_HID);
  __syncthreads();

  const int wave = (blockIdx.x * blockDim.x + threadIdx.x) >> 5;
  const int row0 = wave * 16;
  if (row0 >= nrows) return;
  const int lane = threadIdx.x & 31;
  int row = row0 + (lane & 15);
  if (row >= nrows) row = nrows - 1;
  const float inv = 1.0f / fmaxf(deg[row], 1.0f);  // per-row scale folded into A

  v8f acc[8];
#pragma unroll
  for (int i = 0; i < 8; ++i)
#pragma unroll
    for (int j = 0; j < 8; ++j) acc[i][j] = 0.0f;
  gemm_strip_accum(agg, D_HID, row0, nrows, D_HID, lw,                       inv,  acc);
  gemm_strip_accum(h,   D_HID, row0, nrows, D_HID, lw + (D_HID / 2) * D_HID, 1.0f, acc);
  store_strip_relu(hout, D_HID, row0, nrows, bl, acc);
}

// one wave per edge: gather h[src] (float4/lane), atomic-add into agg[dst]
__global__ void edge_scatter_kernel(const float* __restrict__ h,
                                    const int* __restrict__ src,
                                    const int* __restrict__ dst,
                                    float* __restrict__ agg, int E)
{
  const int wave = (blockIdx.x * blockDim.x + threadIdx.x) >> 5;
  if (wave >= E) return;
  const int lane = threadIdx.x & 31;
  const int s = src[wave];
  const int d = dst[wave];
  const float4 v = ((const float4*)(h + (size_t)s * D_HID))[lane];
  float* ap = agg + (size_t)d * D_HID + lane * 4;
  atomicAdd(ap + 0, v.x);
  atomicAdd(ap + 1, v.y);
  atomicAdd(ap + 2, v.z);
  atomicAdd(ap + 3, v.w);
}

// monotone-bits float atomic max (init must be -inf)
__device__ __forceinline__ void atomicMaxF(float* addr, float val) {
  if (val >= 0.0f) atomicMax((int*)addr, __float_as_int(val));
  else             atomicMin((unsigned int*)addr, (unsigned int)__float_as_int(val));
}

// one wave per node: accumulate per-graph sum / max / count
__global__ void pool_kernel(const float* __restrict__ h,
                            const int* __restrict__ bidx,
                            float* __restrict__ sum_p,
                            float* __restrict__ max_p,
                            float* __restrict__ cnt, int N)
{
  const int wave = (blockIdx.x * blockDim.x + threadIdx.x) >> 5;
  if (wave >= N) return;
  const int lane = threadIdx.x & 31;
  const int b = bidx[wave];
  const float4 v = ((const float4*)(h + (size_t)wave * D_HID))[lane];
  float* sp = sum_p + (size_t)b * D_HID + lane * 4;
  atomicAdd(sp + 0, v.x);
  atomicAdd(sp + 1, v.y);
  atomicAdd(sp + 2, v.z);
  atomicAdd(sp + 3, v.w);
  float* mp = max_p + (size_t)b * D_HID + lane * 4;
  atomicMaxF(mp + 0, v.x);
  atomicMaxF(mp + 1, v.y);
  atomicMaxF(mp + 2, v.z);
  atomicMaxF(mp + 3, v.w);
  if (lane == 0) atomicAdd(&cnt[b], 1.0f);
}

// z[g] = [ sum/cnt (128) | max (128) | graph_feat (32) ]
__global__ void finalize_z_kernel(const float* __restrict__ sum_p,
                                  const float* __restrict__ max_p,
                                  const float* __restrict__ cnt,
                                  const float* __restrict__ gf,
                                  float* __restrict__ z, int G)
{
  const int idx = blockIdx.x * blockDim.x + threadIdx.x;
  if (idx >= G * Z_DIM) return;
  const int g = idx / Z_DIM;
  const int c = idx % Z_DIM;
  float v;
  if (c < D_HID)           v = sum_p[(size_t)g * D_HID + c] / fmaxf(cnt[g], 1.0f);
  else if (c < 2 * D_HID)  v = max_p[(size_t)g * D_HID + (c - D_HID)];
  else                     v = gf[(size_t)g * D_GFT + (c - 2 * D_HID)];
  z[idx] = v;
}

// ---------------- launch ----------------------------------------------------

static inline int cdiv(long a, long b) { return (int)((a + b - 1) / b); }

extern "C" void kernel_launch(void* const* d_in, const int* in_sizes, int n_in,
                              void* d_out, int out_size, void* d_ws, size_t ws_size,
                              hipStream_t stream)
{
  const float* x    = (const float*)d_in[0];
  const int*   ei   = (const int*)  d_in[1];
  const int*   bidx = (const int*)  d_in[2];
  const float* gf   = (const float*)d_in[3];
  const float* W_in = (const float*)d_in[4];
  const float* b_in = (const float*)d_in[5];
  const float* Wl   = (const float*)d_in[6];
  const float* bl   = (const float*)d_in[7];
  const float* Wr   = (const float*)d_in[8];
  const float* W1   = (const float*)d_in[9];
  const float* b1   = (const float*)d_in[10];
  const float* W2   = (const float*)d_in[11];
  const float* b2   = (const float*)d_in[12];

  const int N    = in_sizes[2];              // 100000 nodes
  const int E    = in_sizes[1] / 2;          // 1.6M edges
  const int G    = in_sizes[3] / D_GFT;      // 512 graphs
  const int D_IN = in_sizes[0] / N;          // 128

  const int* src = ei;
  const int* dst = ei + E;

  // workspace partition (floats)
  float* p = (float*)d_ws;
  float* hA    = p; p += (size_t)N * D_HID;
  float* hB    = p; p += (size_t)N * D_HID;
  float* agg   = p; p += (size_t)N * D_HID;
  float* deg   = p; p += N;
  float* sum_p = p; p += (size_t)G * D_HID;
  float* max_p = p; p += (size_t)G * D_HID;
  float* cnt   = p; p += G;
  float* z     = p; p += (size_t)G * Z_DIM;
  float* t     = p; p += (size_t)G * D_HID;

  const int BT = 256;                         // 8 waves / block
  const int node_strips  = cdiv(N, 16);
  const int node_blocks  = cdiv(node_strips, 8);
  const int graph_strips = cdiv(G, 16);
  const int graph_blocks = cdiv(graph_strips, 8);

  // degrees (dst-only -> identical across layers, compute once)
  fill_kernel<<<cdiv(N, BT), BT, 0, stream>>>(deg, 0.0f, (size_t)N);
  deg_kernel<<<cdiv(E, BT), BT, 0, stream>>>(dst, deg, E);

  // h = relu(x @ W_in + b_in)
  gemm_relu_kernel<D_HID><<<node_blocks, BT, 0, stream>>>(x, D_IN, W_in, b_in,
                                                          hA, D_HID, N);

  // 3 x SAGEConv
  for (int l = 0; l < NLAYER; ++l) {
    fill_kernel<<<cdiv((long)N * D_HID, BT), BT, 0, stream>>>(agg, 0.0f,
                                                              (size_t)N * D_HID);
    edge_scatter_kernel<<<cdiv(E, 8), BT, 0, stream>>>(hA, src, dst, agg, E);
    sage_combine_kernel<<<node_blocks, BT, 0, stream>>>(
        hA, agg, deg,
        Wl + (size_t)l * D_HID * D_HID, bl + (size_t)l * D_HID,
        Wr + (size_t)l * D_HID * D_HID, hB, N);
    float* tmp = hA; hA = hB; hB = tmp;
  }

  // global mean / max pooling
  fill_kernel<<<cdiv((long)G * D_HID, BT), BT, 0, stream>>>(sum_p, 0.0f,
                                                            (size_t)G * D_HID);
  fill_kernel<<<cdiv((long)G * D_HID, BT), BT, 0, stream>>>(max_p, -INFINITY,
                                                            (size_t)G * D_HID);
  fill_kernel<<<cdiv(G, BT), BT, 0, stream>>>(cnt, 0.0f, (size_t)G);
  pool_kernel<<<cdiv(N, 8), BT, 0, stream>>>(hA, bidx, sum_p, max_p, cnt, N);
  finalize_z_kernel<<<cdiv((long)G * Z_DIM, BT), BT, 0, stream>>>(sum_p, max_p,
                                                                  cnt, gf, z, G);

  // MLP head: out = relu(relu(z @ W1 + b1) @ W2 + b2)
  gemm_relu_kernel<Z_DIM><<<graph_blocks, BT, 0, stream>>>(z, Z_DIM, W1, b1,
                                                           t, D_HID, G);
  gemm_relu_kernel<D_HID><<<graph_blocks, BT, 0, stream>>>(t, D_HID, W2, b2,
                                                           (float*)d_out, D_HID, G);
}